// HypNO_PINN_39994735460754
// MI455X (gfx1250) — compile-verified
//
#include <hip/hip_runtime.h>
#include <hip/hip_bf16.h>
#include <math.h>

// ---------------------------------------------------------------------------
// HypNO-PINN forward for MI455X (gfx1250).
// Sizes: B=1, NT=64, NX=512, D=64, H=128, K_LIFT=3, K_X=3, K_T=2, N_LAYERS=1.
// Exploits the t-broadcast of h0 (single MP layer): shock and sp_agg are
// computed per-x only; the (t,x) grid work (tp/upd/residual/dec) runs on
// fp32 WMMA (v_wmma_f32_16x16x4_f32), one 16-point tile per wave32, with the
// h0 tile staged into LDS via the CDNA5 async DMA path
// (global_load_async_to_lds_b128 + s_wait_asynccnt).
// ---------------------------------------------------------------------------

#define NXC 512
#define NTC 64
#define DC  64
#define HC  128
#define THRESH_C 0.1f
#define DELTA_C  0.01f

typedef __attribute__((ext_vector_type(2))) float v2f;
typedef __attribute__((ext_vector_type(8))) float v8f;

__device__ __forceinline__ float gelu_f(float v) {
    // exact GELU: x * 0.5 * (1 + erf(x / sqrt(2)))
    return 0.5f * v * (1.0f + erff(v * 0.70710678118654752440f));
}
__device__ __forceinline__ int clampi(int v, int lo, int hi) {
    return v < lo ? lo : (v > hi ? hi : v);
}

// ---------------------------------------------------------------------------
// Wave-level fp32 GEMM layer: O(16xN) = act( A(16xK) @ W(KxN) + bias ).
// A in LDS row-major (stride K). W,bias in global (L2-hot). O to LDS.
// Fragment layouts per CDNA5 ISA 7.12.2:
//   A 16x4 f32: lanes 0-15 = rows M, VGPR0/1 = K,K+1; lanes 16-31 = K+2,K+3.
//   B 4x16 f32: lanes = cols N, upper half-wave = K+2,K+3.
//   C/D: VGPR r -> M = r (+8 for upper half-wave), lane%16 = N.
// K must be a multiple of 4 (132/128/64 all are).
// ---------------------------------------------------------------------------
__device__ __forceinline__ void gemm16(const float* Alds, int K,
                                       const float* __restrict__ W,
                                       const float* __restrict__ bias, int N,
                                       float* Olds, bool act, int lane) {
    const int row   = lane & 15;
    const int kh    = (lane >> 4) << 1;  // 0 or 2
    const int mbase = (lane >> 4) << 3;  // 0 or 8
    for (int n0 = 0; n0 < N; n0 += 16) {
        v8f acc = {0.f, 0.f, 0.f, 0.f, 0.f, 0.f, 0.f, 0.f};
        for (int k0 = 0; k0 < K; k0 += 4) {
            v2f a, b;
            a.x = Alds[row * K + k0 + kh];
            a.y = Alds[row * K + k0 + kh + 1];
            b.x = W[(k0 + kh) * N + n0 + row];
            b.y = W[(k0 + kh + 1) * N + n0 + row];
            acc = __builtin_amdgcn_wmma_f32_16x16x4_f32(false, a, false, b,
                                                        (short)0, acc, false, false);
        }
#pragma unroll
        for (int r = 0; r < 8; ++r) {
            const int m = mbase + r;
            const int n = n0 + row;
            float v = acc[r] + bias[n];
            Olds[m * N + n] = act ? gelu_f(v) : v;
        }
    }
}

// Fragment-only variant: returns the 16x16 C tile for column tile n0 (no bias).
__device__ __forceinline__ v8f gemm16_frag(const float* Alds, int K,
                                           const float* __restrict__ W, int N,
                                           int n0, int lane) {
    const int row = lane & 15;
    const int kh  = (lane >> 4) << 1;
    v8f acc = {0.f, 0.f, 0.f, 0.f, 0.f, 0.f, 0.f, 0.f};
    for (int k0 = 0; k0 < K; k0 += 4) {
        v2f a, b;
        a.x = Alds[row * K + k0 + kh];
        a.y = Alds[row * K + k0 + kh + 1];
        b.x = W[(k0 + kh) * N + n0 + row];
        b.y = W[(k0 + kh + 1) * N + n0 + row];
        acc = __builtin_amdgcn_wmma_f32_16x16x4_f32(false, a, false, b,
                                                    (short)0, acc, false, false);
    }
    return acc;
}

// ---------------------------------------------------------------------------
// Kernel 1: lift. One block (128 thr) per x point. h0[x][0..63].
// ---------------------------------------------------------------------------
__global__ __launch_bounds__(128) void lift_kernel(
    const float* __restrict__ u0, const float* __restrict__ x,
    const float* __restrict__ Wn0, const float* __restrict__ bn0,
    const float* __restrict__ Wn1, const float* __restrict__ bn1,
    const float* __restrict__ We0, const float* __restrict__ be0,
    const float* __restrict__ We1, const float* __restrict__ be1,
    const float* __restrict__ Wg0, const float* __restrict__ bg0,
    const float* __restrict__ Wg1, const float* __restrict__ bg1,
    const float* __restrict__ Wc0, const float* __restrict__ bc0,
    const float* __restrict__ Wc1, const float* __restrict__ bc1,
    float* __restrict__ h0) {
    const int xi = blockIdx.x;
    const int tid = threadIdx.x;
    __shared__ float hid[HC], hidG[HC], hs[DC], aggv[DC], msgv[DC], cin[HC], red[HC];

    const float u0v = u0[xi], xv = x[xi];
    // node MLP [2 -> 128 -> 64]
    hid[tid] = gelu_f(u0v * Wn0[tid] + xv * Wn0[HC + tid] + bn0[tid]);
    __syncthreads();
    if (tid < DC) {
        float s = bn1[tid];
        for (int k = 0; k < HC; ++k) s += hid[k] * Wn1[k * DC + tid];
        hs[tid] = s;
        aggv[tid] = 0.f;
    }
    __syncthreads();

    for (int j = -3; j <= 3; ++j) {
        const int xj = clampi(xi + j, 0, NXC - 1);
        const float uj = u0[xj];
        const float dxv = x[xj] - xv;
        const float e3 = fabsf(u0v - uj);
        // edge + gate hidden layers [4 -> 128]
        float s = u0v * We0[tid] + uj * We0[HC + tid] + dxv * We0[2 * HC + tid] +
                  e3 * We0[3 * HC + tid] + be0[tid];
        hid[tid] = gelu_f(s);
        s = u0v * Wg0[tid] + uj * Wg0[HC + tid] + dxv * Wg0[2 * HC + tid] +
            e3 * Wg0[3 * HC + tid] + bg0[tid];
        hidG[tid] = gelu_f(s);
        __syncthreads();
        if (tid < DC) {
            float m = be1[tid];
            for (int k = 0; k < HC; ++k) m += hid[k] * We1[k * DC + tid];
            msgv[tid] = m;
        }
        red[tid] = hidG[tid] * Wg1[tid];  // Wg1 is 128x1
        __syncthreads();
        for (int off = 64; off > 0; off >>= 1) {
            if (tid < off) red[tid] += red[tid + off];
            __syncthreads();
        }
        const float gate = 1.0f / (1.0f + expf(-(red[0] + bg1[0])));
        if (tid < DC) aggv[tid] += gate * msgv[tid];
        __syncthreads();
    }
    // comb MLP [128 -> 128 -> 64]
    if (tid < DC) { cin[tid] = hs[tid]; cin[DC + tid] = aggv[tid]; }
    __syncthreads();
    {
        float s = bc0[tid];
        for (int k = 0; k < HC; ++k) s += cin[k] * Wc0[k * HC + tid];
        hid[tid] = gelu_f(s);
    }
    __syncthreads();
    if (tid < DC) {
        float s = bc1[tid];
        for (int k = 0; k < HC; ++k) s += hid[k] * Wc1[k * DC + tid];
        h0[xi * DC + tid] = s;
    }
}

// ---------------------------------------------------------------------------
// Kernel 2: shock decoder u[x] = dec(h0[x]) (t-independent since h is a
// t-broadcast of h0). One block (128 thr) per x.
// ---------------------------------------------------------------------------
__global__ __launch_bounds__(128) void ucoarse_kernel(
    const float* __restrict__ h0,
    const float* __restrict__ Ws0, const float* __restrict__ bs0,
    const float* __restrict__ Ws1, const float* __restrict__ bs1,
    float* __restrict__ uarr) {
    const int xi = blockIdx.x;
    const int tid = threadIdx.x;
    __shared__ float hid[HC], red[HC];
    float s = bs0[tid];
    for (int k = 0; k < DC; ++k) s += h0[xi * DC + k] * Ws0[k * HC + tid];
    hid[tid] = gelu_f(s);
    __syncthreads();
    red[tid] = hid[tid] * Ws1[tid];
    __syncthreads();
    for (int off = 64; off > 0; off >>= 1) {
        if (tid < off) red[tid] += red[tid + off];
        __syncthreads();
    }
    if (tid == 0) uarr[xi] = red[0] + bs1[0];
}

// ---------------------------------------------------------------------------
// Kernel 3: shock indicator. du/dt == 0 (u constant in t), so
// r[x] = |df/dx|, shock[x] = r / max(max_x r, 1e-8). One block, 512 thr.
// ---------------------------------------------------------------------------
__global__ __launch_bounds__(512) void shock_kernel(
    const float* __restrict__ x, const float* __restrict__ uarr,
    float* __restrict__ shockv) {
    const int i = threadIdx.x;
    __shared__ float f[NXC], red[NXC];
    const float u = uarr[i];
    f[i] = u * (1.0f - u);
    __syncthreads();
    const float dx = x[1] - x[0];
    float df;
    if (i == 0)            df = (f[1] - f[0]) / dx;
    else if (i == NXC - 1) df = (f[NXC - 1] - f[NXC - 2]) / dx;
    else                   df = (f[i + 1] - f[i - 1]) / (2.0f * dx);
    const float r = fabsf(df);
    red[i] = r;
    __syncthreads();
    for (int off = NXC / 2; off > 0; off >>= 1) {
        if (i < off) red[i] = fmaxf(red[i], red[i + off]);
        __syncthreads();
    }
    const float rmax = fmaxf(red[0], 1e-8f);
    shockv[i] = r / rmax;
}

// ---------------------------------------------------------------------------
// Kernel 4: spatial aggregation (t-independent — msg_in has no t terms when
// h is a t-broadcast). One block (128 thr) per x; 7 neighbors,
// sp MLP [133 -> 128 -> 128 -> 64], shock-gated clip.
// ---------------------------------------------------------------------------
__global__ __launch_bounds__(128) void spagg_kernel(
    const float* __restrict__ u0, const float* __restrict__ x,
    const float* __restrict__ h0, const float* __restrict__ shockv,
    const float* __restrict__ Ws0, const float* __restrict__ bs0,
    const float* __restrict__ Ws1, const float* __restrict__ bs1,
    const float* __restrict__ Ws2, const float* __restrict__ bs2,
    float* __restrict__ spagg) {
    const int xi = blockIdx.x;
    const int tid = threadIdx.x;
    __shared__ float ein[133], hid1[HC], hid2[HC];
    const bool is_shock = shockv[xi] > THRESH_C;
    float acc = 0.f;
    if (tid < DC) ein[tid] = h0[xi * DC + tid];
    for (int j = -3; j <= 3; ++j) {
        const int xj = clampi(xi + j, 0, NXC - 1);
        if (tid < DC) ein[DC + tid] = h0[xj * DC + tid];
        if (tid == 0) {
            ein[128] = x[xi];
            ein[129] = x[xj];
            ein[130] = x[xj] - x[xi];
            const float du0 = u0[xi] - u0[xj];
            ein[131] = du0;
            ein[132] = fabsf(du0);
        }
        __syncthreads();
        float s = bs0[tid];
        for (int k = 0; k < 133; ++k) s += ein[k] * Ws0[k * HC + tid];
        hid1[tid] = gelu_f(s);
        __syncthreads();
        s = bs1[tid];
        for (int k = 0; k < HC; ++k) s += hid1[k] * Ws1[k * HC + tid];
        hid2[tid] = gelu_f(s);
        __syncthreads();
        if (tid < DC) {
            float m = bs2[tid];
            for (int k = 0; k < HC; ++k) m += hid2[k] * Ws2[k * DC + tid];
            if (is_shock) m = fminf(fmaxf(m, 0.f), DELTA_C);
            acc += m;
        }
        __syncthreads();
    }
    if (tid < DC) spagg[xi * DC + tid] = acc;
}

// ---------------------------------------------------------------------------
// Kernel 5 (WMMA + async DMA): temporal messages + update MLP + residual +
// decoder over the full (t,x) grid. One wave32 per 16-point tile (constant t,
// contiguous x). The 16x64 h0 tile is DMA'd into per-wave LDS once via
// global_load_async_to_lds_b128 (ASYNCcnt), then consumed by all five
// GEMM stages. 2 waves / block -> 41.5 KB static LDS.
// ---------------------------------------------------------------------------
#define MPW 2  // waves per block
#define WAVE_LDS (16 * 132 + 16 * 128 + 16 * 64)
__global__ __launch_bounds__(32 * MPW) void mp_dec_kernel(
    const float* __restrict__ xg, const float* __restrict__ tg,
    const float* __restrict__ h0, const float* __restrict__ spagg,
    const float* __restrict__ shockv,
    const float* __restrict__ Wt0, const float* __restrict__ bt0,
    const float* __restrict__ Wt1, const float* __restrict__ bt1,
    const float* __restrict__ Wt2, const float* __restrict__ bt2,
    const float* __restrict__ Wu0, const float* __restrict__ bu0,
    const float* __restrict__ Wu1, const float* __restrict__ bu1,
    const float* __restrict__ Wu2, const float* __restrict__ bu2,
    const float* __restrict__ Wl,  const float* __restrict__ bl,
    const float* __restrict__ Wd0, const float* __restrict__ bd0,
    const float* __restrict__ Wd1, const float* __restrict__ bd1,
    float* __restrict__ out) {
    __shared__ float smem[MPW * WAVE_LDS];
    const int wid  = threadIdx.x >> 5;
    const int lane = threadIdx.x & 31;
    float* A   = smem + wid * WAVE_LDS;   // activations (<=16x132)
    float* Hb  = A + 16 * 132;            // hidden buffer (16x128)
    float* H0s = Hb + 16 * 128;           // staged h0 tile (16x64)

    const int tile = blockIdx.x * MPW + wid;
    const int p0 = tile * 16;        // 16 | NX -> tile never straddles a t-row
    const int tt = p0 >> 9;          // t index
    const int x0 = p0 & (NXC - 1);   // x base
    const float tI = tg[tt];
    const float inv_t = 1.0f / fmaxf(tI, 1e-6f);

    const int row   = lane & 15;
    const int kh    = (lane >> 4) << 1;
    const int mbase = (lane >> 4) << 3;

    __builtin_prefetch(Wt0, 0, 3);  // global_prefetch_b8: warm the big weight

    // ---- async DMA: stage the 16x64 h0 tile (4 KB) into LDS -------------
    // GVS mode: mem = SADDR(sgpr pair) + VADDR(u32 offset); VDST = LDS addr.
    // Low 32 bits of a generic __shared__ pointer are the LDS byte address.
    {
        const unsigned ldsbase = (unsigned)(uintptr_t)H0s;
        const float* gbase = h0 + (size_t)x0 * DC;   // wave-uniform -> SGPRs
#pragma unroll
        for (int it = 0; it < 8; ++it) {
            const unsigned off  = (unsigned)((it * 32 + lane) * 16);  // bytes
            const unsigned ldsa = ldsbase + off;
            asm volatile("global_load_async_to_lds_b128 %0, %1, %2 offset:0"
                         :: "v"(ldsa), "v"(off), "s"(gbase) : "memory");
        }
        asm volatile("s_wait_asynccnt 0x0" ::: "memory");
    }

    v8f tpagg[4];
#pragma unroll
    for (int nt = 0; nt < 4; ++nt)
#pragma unroll
        for (int r = 0; r < 8; ++r) tpagg[nt][r] = 0.f;

    // ---- temporal messages: j in {-2,-1,0}; tp MLP [132->128->128->64] ----
    for (int j = -2; j <= 0; ++j) {
        const float tJ = tg[clampi(tt + j, 0, NTC - 1)];
        // build msg_in rows (16 x 132) into A; h and h_j are both h0(x)
        for (int idx = lane; idx < 16 * 132; idx += 32) {
            const int r_ = idx / 132, c = idx % 132;
            float v;
            if (c < 64)       v = H0s[r_ * DC + c];
            else if (c < 128) v = H0s[r_ * DC + (c - 64)];
            else if (c == 128) v = tI;
            else if (c == 129) v = tJ;
            else if (c == 130) v = tJ - tI;
            else               v = xg[x0 + r_] * inv_t;
            A[r_ * 132 + c] = v;
        }
        __syncthreads();
        gemm16(A, 132, Wt0, bt0, 128, Hb, true, lane);
        __syncthreads();
        gemm16(Hb, 128, Wt1, bt1, 128, A, true, lane);
        __syncthreads();
#pragma unroll
        for (int nt = 0; nt < 4; ++nt) {
            v8f acc = gemm16_frag(A, 128, Wt2, 64, nt * 16, lane);
            const int n = nt * 16 + row;
#pragma unroll
            for (int r = 0; r < 8; ++r) {
                const int m = mbase + r;
                float v = acc[r] + bt2[n];
                if (shockv[x0 + m] > THRESH_C) v = fminf(fmaxf(v, 0.f), DELTA_C);
                tpagg[nt][r] += v;
            }
        }
        __syncthreads();
    }

    // ---- update MLP input: [h0 | spagg + tpagg] (16 x 128) ----
    for (int idx = lane; idx < 16 * 64; idx += 32) {
        const int r_ = idx >> 6, c = idx & 63;
        A[r_ * 128 + c] = H0s[r_ * DC + c];
    }
#pragma unroll
    for (int nt = 0; nt < 4; ++nt) {
        const int n = nt * 16 + row;
#pragma unroll
        for (int r = 0; r < 8; ++r) {
            const int m = mbase + r;
            A[m * 128 + 64 + n] = tpagg[nt][r] + spagg[(x0 + m) * DC + n];
        }
    }
    __syncthreads();
    gemm16(A, 128, Wu0, bu0, 128, Hb, true, lane);
    __syncthreads();
    gemm16(Hb, 128, Wu1, bu1, 128, A, true, lane);
    __syncthreads();

    // ---- h_nl + residual (h0 @ Wl + bl), then GELU -> h1 ----
    v8f hfrag[4];
#pragma unroll
    for (int nt = 0; nt < 4; ++nt) {
        v8f acc = gemm16_frag(A, 128, Wu2, 64, nt * 16, lane);
        // residual GEMM: A-fragments from the staged h0 tile (16x64 in LDS)
        v8f racc = {0.f, 0.f, 0.f, 0.f, 0.f, 0.f, 0.f, 0.f};
        for (int k0 = 0; k0 < 64; k0 += 4) {
            v2f a, b;
            a.x = H0s[row * DC + k0 + kh];
            a.y = H0s[row * DC + k0 + kh + 1];
            b.x = Wl[(k0 + kh) * DC + nt * 16 + row];
            b.y = Wl[(k0 + kh + 1) * DC + nt * 16 + row];
            racc = __builtin_amdgcn_wmma_f32_16x16x4_f32(false, a, false, b,
                                                         (short)0, racc, false, false);
        }
        const int n = nt * 16 + row;
#pragma unroll
        for (int r = 0; r < 8; ++r)
            hfrag[nt][r] = gelu_f(acc[r] + bu2[n] + racc[r] + bl[n]);
    }
    __syncthreads();
    // store h1 (16 x 64) row-major into A for the decoder
#pragma unroll
    for (int nt = 0; nt < 4; ++nt) {
        const int n = nt * 16 + row;
#pragma unroll
        for (int r = 0; r < 8; ++r) A[(mbase + r) * 64 + n] = hfrag[nt][r];
    }
    __syncthreads();

    // ---- decoder: [64 -> 128 (gelu) -> 1] ----
    gemm16(A, 64, Wd0, bd0, 128, Hb, true, lane);
    __syncthreads();
    {
        const int p = row;
        const int half = (lane >> 4) * 64;
        float s = 0.f;
        for (int k = 0; k < 64; ++k) s += Hb[p * 128 + half + k] * Wd1[half + k];
        const float s2 = __shfl_xor(s, 16, 32);
        if (lane < 16) out[p0 + p] = s + s2 + bd1[0];
    }
}

// ---------------------------------------------------------------------------
// Launch. d_in is the flattened setup_inputs() pytree in insertion order.
// ---------------------------------------------------------------------------
extern "C" void kernel_launch(void* const* d_in, const int* in_sizes, int n_in,
                              void* d_out, int out_size, void* d_ws, size_t ws_size,
                              hipStream_t stream) {
    (void)in_sizes; (void)n_in; (void)out_size; (void)ws_size;
    const float* u0 = (const float*)d_in[0];
    const float* x  = (const float*)d_in[1];
    const float* t  = (const float*)d_in[2];
    // lift: node, edge, gate, comb
    const float* Wn0 = (const float*)d_in[3];  const float* bn0 = (const float*)d_in[4];
    const float* Wn1 = (const float*)d_in[5];  const float* bn1 = (const float*)d_in[6];
    const float* We0 = (const float*)d_in[7];  const float* be0 = (const float*)d_in[8];
    const float* We1 = (const float*)d_in[9];  const float* be1 = (const float*)d_in[10];
    const float* Wg0 = (const float*)d_in[11]; const float* bg0 = (const float*)d_in[12];
    const float* Wg1 = (const float*)d_in[13]; const float* bg1 = (const float*)d_in[14];
    const float* Wc0 = (const float*)d_in[15]; const float* bc0 = (const float*)d_in[16];
    const float* Wc1 = (const float*)d_in[17]; const float* bc1 = (const float*)d_in[18];
    // shock dec
    const float* Wsd0 = (const float*)d_in[19]; const float* bsd0 = (const float*)d_in[20];
    const float* Wsd1 = (const float*)d_in[21]; const float* bsd1 = (const float*)d_in[22];
    // mp layer 0: sp, tp, upd, W
    const float* Wsp0 = (const float*)d_in[23]; const float* bsp0 = (const float*)d_in[24];
    const float* Wsp1 = (const float*)d_in[25]; const float* bsp1 = (const float*)d_in[26];
    const float* Wsp2 = (const float*)d_in[27]; const float* bsp2 = (const float*)d_in[28];
    const float* Wt0  = (const float*)d_in[29]; const float* bt0  = (const float*)d_in[30];
    const float* Wt1  = (const float*)d_in[31]; const float* bt1  = (const float*)d_in[32];
    const float* Wt2  = (const float*)d_in[33]; const float* bt2  = (const float*)d_in[34];
    const float* Wu0  = (const float*)d_in[35]; const float* bu0  = (const float*)d_in[36];
    const float* Wu1  = (const float*)d_in[37]; const float* bu1  = (const float*)d_in[38];
    const float* Wu2  = (const float*)d_in[39]; const float* bu2  = (const float*)d_in[40];
    const float* Wl   = (const float*)d_in[41]; const float* bl   = (const float*)d_in[42];
    // final decoder
    const float* Wd0 = (const float*)d_in[43]; const float* bd0 = (const float*)d_in[44];
    const float* Wd1 = (const float*)d_in[45]; const float* bd1 = (const float*)d_in[46];

    float* ws     = (float*)d_ws;
    float* h0     = ws;              // 512*64
    float* uarr   = ws + 32768;      // 512
    float* shockv = ws + 33280;      // 512
    float* spagg  = ws + 33792;      // 512*64
    float* outp   = (float*)d_out;   // 64*512

    lift_kernel<<<NXC, 128, 0, stream>>>(u0, x, Wn0, bn0, Wn1, bn1, We0, be0, We1, be1,
                                         Wg0, bg0, Wg1, bg1, Wc0, bc0, Wc1, bc1, h0);
    ucoarse_kernel<<<NXC, 128, 0, stream>>>(h0, Wsd0, bsd0, Wsd1, bsd1, uarr);
    shock_kernel<<<1, NXC, 0, stream>>>(x, uarr, shockv);
    spagg_kernel<<<NXC, 128, 0, stream>>>(u0, x, h0, shockv,
                                          Wsp0, bsp0, Wsp1, bsp1, Wsp2, bsp2, spagg);
    const int tiles = (NTC * NXC) / 16;           // 2048
    mp_dec_kernel<<<tiles / MPW, 32 * MPW, 0, stream>>>(
        x, t, h0, spagg, shockv,
        Wt0, bt0, Wt1, bt1, Wt2, bt2,
        Wu0, bu0, Wu1, bu1, Wu2, bu2,
        Wl, bl, Wd0, bd0, Wd1, bd1, outp);
}